// EntropyLoss_57586921505231
// MI455X (gfx1250) — compile-verified
//
#include <hip/hip_runtime.h>
#include <hip/hip_bf16.h>
#include <math.h>

typedef __attribute__((ext_vector_type(2))) float v2f;
typedef __attribute__((ext_vector_type(8))) float v8f;

#define B_DIM 16
#define C_DIM 512
#define D_DIM 4096
#define K_SEL 51                     // C_DIM / 10
#define NSUPER (C_DIM / 32)          // 16 supertiles of 32x32 per dim
#define NPAIR (NSUPER * (NSUPER + 1) / 2)   // 136 triangular pairs

// ---------------------------------------------------------------------------
// 0) zero the three per-tensor accumulators
// ---------------------------------------------------------------------------
__global__ __launch_bounds__(32) void init_sums_kernel(float* __restrict__ sums) {
    if (threadIdx.x < 3) sums[threadIdx.x] = 0.0f;
}

// ---------------------------------------------------------------------------
// 1) per-row squared norms: sq[b*C + c] = sum_d X[b,c,d]^2
// ---------------------------------------------------------------------------
__global__ __launch_bounds__(256) void rownorm_kernel(const float* __restrict__ X,
                                                      float* __restrict__ sq) {
    const int row = blockIdx.x;                       // 0 .. B*C-1
    const float4* xr = (const float4*)(X + (size_t)row * D_DIM);
    float s = 0.0f;
#pragma unroll
    for (int i = 0; i < D_DIM / 4 / 256; ++i) {       // 4 iters of float4
        float4 v = xr[threadIdx.x + 256 * i];
        s += v.x * v.x + v.y * v.y + v.z * v.z + v.w * v.w;
    }
    __shared__ float red[256];
    red[threadIdx.x] = s;
    __syncthreads();
    for (int off = 128; off > 0; off >>= 1) {
        if (threadIdx.x < off) red[threadIdx.x] += red[threadIdx.x + off];
        __syncthreads();
    }
    if (threadIdx.x == 0) sq[row] = red[0];
}

// ---------------------------------------------------------------------------
// 2) symmetric Gram via V_WMMA_F32_16X16X4_F32 with 2x2 register blocking.
//    One wave computes a 32x32 block of G = X_b * X_b^T for supertile pair
//    (st_m <= st_n), converts to distances and mirror-stores both triangles.
//
//    A fragment (16x4 fp32): lane = (k>=2)*16 + m, vgpr = k&1.
//    B fragment (4x16) mirrors it from the column-block rows.
//    C/D layout: VGPR v, lane l -> M = v + 8*(l>=16), N = l%16.
// ---------------------------------------------------------------------------
__device__ __forceinline__ v8f wmma4(v2f a, v2f b, v8f c) {
    // 8 args: (neg_a, A, neg_b, B, c_mod, C, reuse_a, reuse_b)
    return __builtin_amdgcn_wmma_f32_16x16x4_f32(false, a, false, b,
                                                 (short)0, c, false, false);
}

__device__ __forceinline__ void dist_store_tile(const v8f& acc,
                                                const float* __restrict__ sqb,
                                                float* __restrict__ distb,
                                                int tm16, int tn16,
                                                int lm, int half) {
    const int   n   = tn16 + lm;
    const float sqn = sqb[n];
#pragma unroll
    for (int v = 0; v < 8; ++v) {
        const int   m  = tm16 + v + 8 * half;
        const float d2 = sqb[m] + sqn - 2.0f * acc[v];
        const float d  = sqrtf(fmaxf(d2, 0.0f));
        distb[(size_t)m * C_DIM + n] = d;   // upper-triangle tile
        distb[(size_t)n * C_DIM + m] = d;   // mirrored tile (consecutive/lane)
    }
}

__global__ __launch_bounds__(32) void gram_dist_kernel(const float* __restrict__ X,
                                                       const float* __restrict__ sq,
                                                       float* __restrict__ dist) {
    // decode triangular pair index: pair = st_n*(st_n+1)/2 + st_m, st_m <= st_n
    const int pair = blockIdx.x;
    int st_n = 0;
    while ((st_n + 1) * (st_n + 2) / 2 <= pair) ++st_n;
    const int st_m = pair - st_n * (st_n + 1) / 2;

    const int b    = blockIdx.y;
    const int lane = threadIdx.x;
    const int half = lane >> 4;           // K sub-half (0|1)
    const int lm   = lane & 15;

    const float* Xb = X + (size_t)b * C_DIM * D_DIM;
    const float* a0 = Xb + (size_t)(st_m * 32 + lm) * D_DIM + 2 * half;
    const float* a1 = a0 + (size_t)16 * D_DIM;
    const float* b0 = Xb + (size_t)(st_n * 32 + lm) * D_DIM + 2 * half;
    const float* b1 = b0 + (size_t)16 * D_DIM;

    v8f acc00 = {}, acc01 = {}, acc10 = {}, acc11 = {};
    for (int k0 = 0; k0 < D_DIM; k0 += 16) {
#pragma unroll
        for (int kk = 0; kk < 16; kk += 4) {
            const v2f fa0 = *(const v2f*)(a0 + k0 + kk);
            const v2f fa1 = *(const v2f*)(a1 + k0 + kk);
            const v2f fb0 = *(const v2f*)(b0 + k0 + kk);
            const v2f fb1 = *(const v2f*)(b1 + k0 + kk);
            acc00 = wmma4(fa0, fb0, acc00);
            acc01 = wmma4(fa0, fb1, acc01);
            acc10 = wmma4(fa1, fb0, acc10);
            acc11 = wmma4(fa1, fb1, acc11);
        }
    }

    const float* sqb   = sq + b * C_DIM;
    float*       distb = dist + (size_t)b * C_DIM * C_DIM;
    const int tm0 = st_m * 32, tn0 = st_n * 32;
    dist_store_tile(acc00, sqb, distb, tm0,      tn0,      lm, half);
    dist_store_tile(acc01, sqb, distb, tm0,      tn0 + 16, lm, half);
    dist_store_tile(acc10, sqb, distb, tm0 + 16, tn0,      lm, half);
    dist_store_tile(acc11, sqb, distb, tm0 + 16, tn0 + 16, lm, half);
}

// ---------------------------------------------------------------------------
// 3) exact K-th order statistic per row (512 values), accumulate into sums[t]
// ---------------------------------------------------------------------------
__global__ __launch_bounds__(256) void kth_select_kernel(const float* __restrict__ dist,
                                                         float* __restrict__ sums,
                                                         int t) {
    const int row = blockIdx.x;                       // 0 .. B*C-1
    __shared__ float vals[C_DIM];
    __shared__ float rball;
    const float* dr = dist + (size_t)row * C_DIM;
    vals[threadIdx.x]       = dr[threadIdx.x];
    vals[threadIdx.x + 256] = dr[threadIdx.x + 256];
    __syncthreads();
#pragma unroll
    for (int c = 0; c < 2; ++c) {
        const float v = vals[threadIdx.x + 256 * c];
        int less = 0, eq = 0;
        for (int j = 0; j < C_DIM; ++j) {
            const float w = vals[j];
            less += (w < v);
            eq   += (w == v);
        }
        // unique order statistic: exactly one distinct value satisfies this
        if (less <= K_SEL && K_SEL < less + eq) rball = v;
    }
    __syncthreads();
    if (threadIdx.x == 0) atomicAdd(&sums[t], rball);
}

// ---------------------------------------------------------------------------
// 4) finalize: e_t = log(sum_t + 1); out = var([e1-e0, e2-e1], ddof=1)
// ---------------------------------------------------------------------------
__global__ void finalize_kernel(const float* __restrict__ sums,
                                float* __restrict__ out) {
    const float e0 = logf(sums[0] + 1.0f);
    const float e1 = logf(sums[1] + 1.0f);
    const float e2 = logf(sums[2] + 1.0f);
    const float d1 = e1 - e0;
    const float d2 = e2 - e1;
    const float diff = d1 - d2;
    out[0] = 0.5f * diff * diff;
}

// ---------------------------------------------------------------------------
extern "C" void kernel_launch(void* const* d_in, const int* in_sizes, int n_in,
                              void* d_out, int out_size, void* d_ws, size_t ws_size,
                              hipStream_t stream) {
    const float* feats[3] = {(const float*)d_in[0],
                             (const float*)d_in[1],
                             (const float*)d_in[2]};

    // workspace layout
    float* sums = (float*)d_ws;                          // 3 floats
    float* sq   = (float*)((char*)d_ws + 256);           // B*C floats (32 KB)
    float* dist = (float*)((char*)d_ws + 65536);         // B*C*C floats (16 MB)

    init_sums_kernel<<<1, 32, 0, stream>>>(sums);

    for (int t = 0; t < 3; ++t) {
        rownorm_kernel<<<B_DIM * C_DIM, 256, 0, stream>>>(feats[t], sq);
        dim3 g(NPAIR, B_DIM);                            // 136 x 16 wave-blocks
        gram_dist_kernel<<<g, 32, 0, stream>>>(feats[t], sq, dist);
        kth_select_kernel<<<B_DIM * C_DIM, 256, 0, stream>>>(dist, sums, t);
    }

    finalize_kernel<<<1, 1, 0, stream>>>(sums, (float*)d_out);
}